// MANO_Model_27101243638338
// MI455X (gfx1250) — compile-verified
//
#include <hip/hip_runtime.h>
#include <math.h>

typedef float v2f __attribute__((ext_vector_type(2)));
typedef float v8f __attribute__((ext_vector_type(8)));

#define B_    4096
#define V_    778
#define NJ    16
#define NB_   10
#define NP_   135
// K layout: [0..134]=pose_feat, [135]=0, [136..145]=beta, [146..147]=0
#define KPAD  148
#define NCOLS (V_ * 3)            // 2334
#define NPITCH 2336               // padded col count / v_posed pitch
#define NTILE (NPITCH / 16)       // 146
#define MTILE (B_ / 16)           // 256

// workspace layout (float offsets)
#define WS_JTJS 0                            // 528 floats: JT(48) + JS(480)
#define WS_PF   1024                         // B*148  A' = [pose_feat|beta] padded
#define WS_R    (WS_PF + B_ * KPAD)          // B*16*9 rotation matrices
#define WS_AREL (WS_R + B_ * NJ * 9)         // B*16*12 relative transforms
#define WS_PP   (WS_AREL + B_ * NJ * 12)     // NPITCH*148 padded B' = [P^T|S^T]
#define WS_VP   (WS_PP + NPITCH * KPAD)      // B*NPITCH v_posed

// ---------------------------------------------------------------------------
// K0a: batch-independent JT = J_reg @ T_bar (16x3), JS = J_reg @ S (16x3x10)
// ---------------------------------------------------------------------------
__global__ void k_jreg(const float* __restrict__ J_reg,
                       const float* __restrict__ T_bar,
                       const float* __restrict__ S,
                       float* __restrict__ ws) {
  int t = blockIdx.x * blockDim.x + threadIdx.x;
  if (t >= 528) return;
  float acc = 0.f;
  if (t < 48) {
    int j = t / 3, d = t % 3;
    const float* jr = J_reg + j * V_;
    for (int v = 0; v < V_; ++v) acc += jr[v] * T_bar[v * 3 + d];
  } else {
    int t2 = t - 48;
    int j = t2 / 30, rem = t2 % 30, d = rem / 10, l = rem % 10;
    const float* jr = J_reg + j * V_;
    for (int v = 0; v < V_; ++v) acc += jr[v] * S[(v * 3 + d) * NB_ + l];
  }
  ws[WS_JTJS + t] = acc;
}

// ---------------------------------------------------------------------------
// K0b: build padded B' operand: Ppad[n][k], n in [0,2336), k in [0,148)
//   k<135   -> P[n*135+k]      (n < 2334)
//   k==135  -> 0
//   136..145-> S[n*10 + k-136] (n < 2334)
//   146,147 -> 0
// ---------------------------------------------------------------------------
__global__ void k_padP(const float* __restrict__ P,
                       const float* __restrict__ S,
                       float* __restrict__ ws) {
  int t = blockIdx.x * blockDim.x + threadIdx.x;
  if (t >= NPITCH * KPAD) return;
  int n = t / KPAD, k = t % KPAD;
  float v = 0.0f;
  if (n < NCOLS) {
    if (k < NP_) v = P[(size_t)n * NP_ + k];
    else if (k >= 136 && k < 136 + NB_) v = S[(size_t)n * NB_ + (k - 136)];
  }
  ws[WS_PP + t] = v;
}

// ---------------------------------------------------------------------------
// K1: Rodrigues per (b,j); write R (row-major 9) and A' row
//     (pose_feat entries for j>=1; j==0 thread fills pad + beta slots)
// ---------------------------------------------------------------------------
__global__ void k_rodrigues(const float* __restrict__ pose,
                            const float* __restrict__ beta,
                            float* __restrict__ ws) {
  int t = blockIdx.x * blockDim.x + threadIdx.x;  // b*16 + j
  if (t >= B_ * NJ) return;
  int b = t >> 4, j = t & 15;
  float rx = pose[t * 3 + 0], ry = pose[t * 3 + 1], rz = pose[t * 3 + 2];
  float th = sqrtf(rx * rx + ry * ry + rz * rz + 1e-16f);
  float inv = 1.0f / th;
  float kx = rx * inv, ky = ry * inv, kz = rz * inv;
  float ct = cosf(th), st = sinf(th), oc = 1.0f - ct;
  float R[9];
  R[0] = ct + oc * kx * kx;      R[1] = -st * kz + oc * kx * ky; R[2] = st * ky + oc * kx * kz;
  R[3] = st * kz + oc * kx * ky; R[4] = ct + oc * ky * ky;       R[5] = -st * kx + oc * ky * kz;
  R[6] = -st * ky + oc * kx * kz;R[7] = st * kx + oc * ky * kz;  R[8] = ct + oc * kz * kz;

  float* Rout = ws + WS_R + (size_t)t * 9;
#pragma unroll
  for (int e = 0; e < 9; ++e) Rout[e] = R[e];

  float* pf = ws + WS_PF + (size_t)b * KPAD;
  if (j >= 1) {
    int base = (j - 1) * 9;
#pragma unroll
    for (int e = 0; e < 9; ++e) {
      float iv = (e == 0 || e == 4 || e == 8) ? 1.0f : 0.0f;
      pf[base + e] = R[e] - iv;
    }
  } else {
    pf[135] = 0.0f;
#pragma unroll
    for (int l = 0; l < NB_; ++l) pf[136 + l] = beta[b * NB_ + l];
    pf[146] = 0.0f;
    pf[147] = 0.0f;
  }
}

// ---------------------------------------------------------------------------
// K2: kinematic chain per batch -> posed joints (out) + A_rel (ws)
// PARENTS = [-1, 0,1,2, 0,4,5, 0,7,8, 0,10,11, 0,13,14] = 5 chains of depth 3
// ---------------------------------------------------------------------------
__device__ __forceinline__ void mat3mul(const float* a, const float* b, float* c) {
#pragma unroll
  for (int i = 0; i < 3; ++i)
#pragma unroll
    for (int j = 0; j < 3; ++j)
      c[i * 3 + j] = a[i * 3 + 0] * b[0 * 3 + j] + a[i * 3 + 1] * b[1 * 3 + j] +
                     a[i * 3 + 2] * b[2 * 3 + j];
}
__device__ __forceinline__ void mat3vec(const float* a, const float* v, float* o) {
#pragma unroll
  for (int i = 0; i < 3; ++i)
    o[i] = a[i * 3 + 0] * v[0] + a[i * 3 + 1] * v[1] + a[i * 3 + 2] * v[2];
}

__global__ void k_chain(const float* __restrict__ beta,
                        const float* __restrict__ root_trans,
                        float* __restrict__ ws,
                        float* __restrict__ out_joints) {
  int b = blockIdx.x * blockDim.x + threadIdx.x;
  if (b >= B_) return;

  const float* JT = ws + WS_JTJS;
  const float* JS = ws + WS_JTJS + 48;
  float bet[NB_];
#pragma unroll
  for (int l = 0; l < NB_; ++l) bet[l] = beta[b * NB_ + l];

  float Jb[NJ][3];
#pragma unroll
  for (int j = 0; j < NJ; ++j)
#pragma unroll
    for (int d = 0; d < 3; ++d) {
      float a = JT[j * 3 + d];
#pragma unroll
      for (int l = 0; l < NB_; ++l) a += JS[j * 30 + d * 10 + l] * bet[l];
      Jb[j][d] = a;
    }

  const float* Rb = ws + WS_R + (size_t)b * NJ * 9;
  float* Ab = ws + WS_AREL + (size_t)b * NJ * 12;
  float rt0 = root_trans[b * 3 + 0], rt1 = root_trans[b * 3 + 1], rt2 = root_trans[b * 3 + 2];

  auto emit = [&](int i, const float* rot, const float* tr) {
    out_joints[((size_t)b * 21 + i) * 3 + 0] = tr[0] + rt0;
    out_joints[((size_t)b * 21 + i) * 3 + 1] = tr[1] + rt1;
    out_joints[((size_t)b * 21 + i) * 3 + 2] = tr[2] + rt2;
    float tc[3];
    mat3vec(rot, Jb[i], tc);
    float* o = Ab + i * 12;
#pragma unroll
    for (int r = 0; r < 3; ++r) {
      o[r * 4 + 0] = rot[r * 3 + 0];
      o[r * 4 + 1] = rot[r * 3 + 1];
      o[r * 4 + 2] = rot[r * 3 + 2];
      o[r * 4 + 3] = tr[r] - tc[r];
    }
  };

  float A0r[9], A0t[3];
#pragma unroll
  for (int e = 0; e < 9; ++e) A0r[e] = Rb[e];
  A0t[0] = Jb[0][0]; A0t[1] = Jb[0][1]; A0t[2] = Jb[0][2];
  emit(0, A0r, A0t);

#pragma unroll
  for (int c = 0; c < 5; ++c) {
    float pr[9], pt[3];
#pragma unroll
    for (int e = 0; e < 9; ++e) pr[e] = A0r[e];
    pt[0] = A0t[0]; pt[1] = A0t[1]; pt[2] = A0t[2];
#pragma unroll
    for (int s = 0; s < 3; ++s) {
      int i = 1 + c * 3 + s;
      int par = (s == 0) ? 0 : (i - 1);
      const float* Ri = Rb + i * 9;
      float rel[3] = {Jb[i][0] - Jb[par][0], Jb[i][1] - Jb[par][1], Jb[i][2] - Jb[par][2]};
      float nr[9], nt[3];
      mat3mul(pr, Ri, nr);
      mat3vec(pr, rel, nt);
      nt[0] += pt[0]; nt[1] += pt[1]; nt[2] += pt[2];
      emit(i, nr, nt);
#pragma unroll
      for (int e = 0; e < 9; ++e) pr[e] = nr[e];
      pt[0] = nt[0]; pt[1] = nt[1]; pt[2] = nt[2];
    }
  }
}

// ---------------------------------------------------------------------------
// K3: v_posed = A'(Bx148) @ B'(148xNPITCH) + T_bar epilogue.
// One wave per 16x16 tile; branch-free b64 fragment loads; f32 WMMA over K.
// ---------------------------------------------------------------------------
__global__ void __launch_bounds__(256) k_poseblend(const float* __restrict__ T_bar,
                                                   float* __restrict__ ws) {
  const float* pf = ws + WS_PF;
  const float* pp = ws + WS_PP;
  float* vp = ws + WS_VP;
  int wave = (blockIdx.x * blockDim.x + threadIdx.x) >> 5;
  if (wave >= MTILE * NTILE) return;  // wave-uniform: EXEC all-1s for WMMA
  int lane = threadIdx.x & 31;
  int tm = wave / NTILE, tn = wave % NTILE;
  int m0 = tm * 16, n0 = tn * 16;
  bool lo = lane < 16;
  int idx = lane & 15;
  int arow = m0 + idx;            // A: lanes 0-15 rows M; lanes 16-31 same rows
  int ncol = n0 + idx;            // B: lanes 0-15 cols N; lanes 16-31 same cols
  int koff = lo ? 0 : 2;          // lanes 0-15 hold K=0,1; lanes 16-31 K=2,3

  const float* ap = pf + (size_t)arow * KPAD + koff;   // 8B aligned (even floats)
  const float* bp = pp + (size_t)ncol * KPAD + koff;   // ncol < 2336 always valid
  v8f acc = {};
#pragma unroll 4
  for (int kk = 0; kk < KPAD; kk += 4) {
    v2f a  = *(const v2f*)(ap + kk);
    v2f bf = *(const v2f*)(bp + kk);
    acc = __builtin_amdgcn_wmma_f32_16x16x4_f32(false, a, false, bf, (short)0,
                                                acc, false, false);
  }

  if (ncol < NCOLS) {
    float tb = T_bar[ncol];
#pragma unroll
    for (int r = 0; r < 8; ++r) {
      int row = m0 + r + (lo ? 0 : 8);  // C/D: VGPR r -> M = r (+8 upper half)
      vp[(size_t)row * NPITCH + ncol] = acc[r] + tb;
    }
  }
}

// ---------------------------------------------------------------------------
// K4: LBS skinning. Block = one batch x 256 vertices; A_rel staged in LDS.
// ---------------------------------------------------------------------------
__global__ void __launch_bounds__(256) k_skin(const float* __restrict__ W,
                                              const float* __restrict__ root_trans,
                                              const float* __restrict__ ws,
                                              float* __restrict__ out_mesh,
                                              float* __restrict__ out_joints) {
  __shared__ float sA[NJ * 12];
  __shared__ float sR[3];
  int b = blockIdx.y;
  int tid = threadIdx.x;
  const float* arel = ws + WS_AREL + (size_t)b * NJ * 12;
  if (tid < NJ * 12) sA[tid] = arel[tid];
  if (tid < 3) sR[tid] = root_trans[b * 3 + tid];
  __syncthreads();

  int v = blockIdx.x * 256 + tid;
  if (v >= V_) return;

  const float* vp = ws + WS_VP + (size_t)b * NPITCH + v * 3;
  float px = vp[0], py = vp[1], pz = vp[2];

  float ts[12];
#pragma unroll
  for (int k = 0; k < 12; ++k) ts[k] = 0.0f;
#pragma unroll
  for (int j = 0; j < NJ; ++j) {
    float wj = W[v * NJ + j];
#pragma unroll
    for (int k = 0; k < 12; ++k) ts[k] += wj * sA[j * 12 + k];
  }
  float mx = ts[0] * px + ts[1] * py + ts[2]  * pz + ts[3]  + sR[0];
  float my = ts[4] * px + ts[5] * py + ts[6]  * pz + ts[7]  + sR[1];
  float mz = ts[8] * px + ts[9] * py + ts[10] * pz + ts[11] + sR[2];

  size_t mo = ((size_t)b * V_ + v) * 3;
  out_mesh[mo + 0] = mx;
  out_mesh[mo + 1] = my;
  out_mesh[mo + 2] = mz;

  constexpr int TIPS[5] = {745, 333, 444, 555, 672};
#pragma unroll
  for (int t = 0; t < 5; ++t) {
    if (v == TIPS[t]) {
      size_t jo = ((size_t)b * 21 + 16 + t) * 3;
      out_joints[jo + 0] = mx;
      out_joints[jo + 1] = my;
      out_joints[jo + 2] = mz;
    }
  }
}

// ---------------------------------------------------------------------------
extern "C" void kernel_launch(void* const* d_in, const int* in_sizes, int n_in,
                              void* d_out, int out_size, void* d_ws, size_t ws_size,
                              hipStream_t stream) {
  const float* beta  = (const float*)d_in[0];
  const float* pose  = (const float*)d_in[1];
  const float* root  = (const float*)d_in[2];
  const float* T_bar = (const float*)d_in[3];
  const float* S     = (const float*)d_in[4];
  const float* P     = (const float*)d_in[5];
  const float* J_reg = (const float*)d_in[6];
  const float* W     = (const float*)d_in[7];
  float* out    = (float*)d_out;
  float* mesh   = out;
  float* joints = out + (size_t)B_ * V_ * 3;
  float* ws     = (float*)d_ws;

  k_jreg<<<3, 256, 0, stream>>>(J_reg, T_bar, S, ws);
  k_padP<<<(NPITCH * KPAD + 255) / 256, 256, 0, stream>>>(P, S, ws);
  k_rodrigues<<<(B_ * NJ + 255) / 256, 256, 0, stream>>>(pose, beta, ws);
  k_chain<<<(B_ + 127) / 128, 128, 0, stream>>>(beta, root, ws, joints);
  int tiles = MTILE * NTILE;  // 37376 waves
  k_poseblend<<<(tiles * 32 + 255) / 256, 256, 0, stream>>>(T_bar, ws);
  k_skin<<<dim3((V_ + 255) / 256, B_), 256, 0, stream>>>(W, root, ws, mesh, joints);
}